// SchNet_60361470378648
// MI455X (gfx1250) — compile-verified
//
#include <hip/hip_runtime.h>
#include <hip/hip_bf16.h>
#include <math.h>

typedef __attribute__((ext_vector_type(2))) float v2f;
typedef __attribute__((ext_vector_type(8))) float v8f;

#define HDIM 64
#define LNUM 3
#define TSTR 68            // LDS row stride (floats) for conflict-free A reads
#define BNINV 0.9995003746f // 1/sqrt(1 + 1e-3)

__device__ __forceinline__ float softplusf(float x) {
    return x > 20.0f ? x : log1pf(expf(x));
}

// ---------------------------------------------------------------------------
// prep: w2sum[l][k] = sum_j fW2[l][k][j]; fb2sum[l] = sum_k fb2[l][k]; zero gsum
// ---------------------------------------------------------------------------
__global__ __launch_bounds__(256) void prep_kernel(
    const float* __restrict__ fW2, const float* __restrict__ fb2,
    float* __restrict__ w2sum, float* __restrict__ fb2sum,
    float* __restrict__ gsum)
{
    int t = threadIdx.x;
    if (t < LNUM * HDIM) {
        const float* r = fW2 + (size_t)t * HDIM;
        float s = 0.0f;
        for (int j = 0; j < HDIM; ++j) s += r[j];
        w2sum[t] = s;
    }
    if (t < LNUM) {
        const float* b = fb2 + t * HDIM;
        float s = 0.0f;
        for (int k = 0; k < HDIM; ++k) s += b[k];
        fb2sum[t] = s;
    }
    if (t < HDIM) gsum[t] = 0.0f;
}

// ---------------------------------------------------------------------------
// embedding: h = x @ emb_W + emb_b   ([N,16] @ [16,64])
// ---------------------------------------------------------------------------
__global__ __launch_bounds__(256) void emb_kernel(
    const float* __restrict__ x, const float* __restrict__ emb_W,
    const float* __restrict__ emb_b, float* __restrict__ h, int n)
{
    __shared__ float sW[16 * HDIM];
    __shared__ float sB[HDIM];
    int tid = threadIdx.x;
    for (int i = tid; i < 16 * HDIM; i += 256) sW[i] = emb_W[i];
    if (tid < HDIM) sB[tid] = emb_b[tid];
    __syncthreads();

    int node = blockIdx.x * 256 + tid;
    if (node >= n) return;
    float xv[16];
    const float* xp = x + (size_t)node * 16;
#pragma unroll
    for (int k = 0; k < 16; ++k) xv[k] = xp[k];
    float4* hp = (float4*)(h + (size_t)node * HDIM);
#pragma unroll
    for (int q = 0; q < 16; ++q) {
        float4 o;
        float* po = (float*)&o;
#pragma unroll
        for (int c = 0; c < 4; ++c) {
            int j = q * 4 + c;
            float s = sB[j];
#pragma unroll
            for (int k = 0; k < 16; ++k) s = fmaf(xv[k], sW[k * HDIM + j], s);
            po[c] = s;
        }
        hp[q] = o;
    }
}

// ---------------------------------------------------------------------------
// per-edge filter scalar for all 3 layers:
//   fs[l][e] = cut(e) * ( sum_k tanh(scaled*fW1+fb1)*w2sum + fb2sum )
// ---------------------------------------------------------------------------
__global__ __launch_bounds__(256) void fs_kernel(
    const float* __restrict__ dist,
    const float* __restrict__ fW1, const float* __restrict__ fb1,
    const float* __restrict__ w2sum, const float* __restrict__ fb2sum,
    float* __restrict__ fs, int E)
{
    __shared__ float sA[LNUM * HDIM], sB[LNUM * HDIM], sC[LNUM * HDIM], sD[LNUM];
    int tid = threadIdx.x;
    if (tid < LNUM * HDIM) { sA[tid] = fW1[tid]; sB[tid] = fb1[tid]; sC[tid] = w2sum[tid]; }
    if (tid < LNUM) sD[tid] = fb2sum[tid];
    __syncthreads();

    int e = blockIdx.x * 256 + tid;
    if (e >= E) return;
    float d = dist[e];
    float scaled = d * (2.0f / 5.0f) - 1.0f;
    float cut = (d <= 5.0f) ? 0.5f * (cosf(d * 3.14159265358979f / 5.0f) + 1.0f) : 0.0f;
#pragma unroll
    for (int l = 0; l < LNUM; ++l) {
        const float* a = &sA[l * HDIM];
        const float* b = &sB[l * HDIM];
        const float* c = &sC[l * HDIM];
        float acc = sD[l];
        for (int k = 0; k < HDIM; ++k)
            acc = fmaf(tanhf(fmaf(scaled, a[k], b[k])), c[k], acc);
        fs[(size_t)l * E + e] = acc * cut;
    }
}

// ---------------------------------------------------------------------------
// zero a float buffer
// ---------------------------------------------------------------------------
__global__ void zero_kernel(float* __restrict__ p, long long n) {
    long long i = (long long)blockIdx.x * 256 + threadIdx.x;
    if (i < n) p[i] = 0.0f;
}

// ---------------------------------------------------------------------------
// scatter: agg[row[e]] += fs[e] * h[col[e]]   (16 threads / edge, float4 each)
// ---------------------------------------------------------------------------
__global__ __launch_bounds__(256) void scatter_kernel(
    const int* __restrict__ ei, const float* __restrict__ h,
    const float* __restrict__ fs, float* __restrict__ agg, int E)
{
    int t = blockIdx.x * 256 + threadIdx.x;
    int e = t >> 4;
    if (e >= E) return;
    int c = (t & 15) * 4;
    int row = ei[e];
    int col = ei[E + e];
    float f = fs[e];
    const float4 hv = *(const float4*)(h + (size_t)col * HDIM + c);
    float* ap = agg + (size_t)row * HDIM + c;
    atomicAdd(ap + 0, hv.x * f);
    atomicAdd(ap + 1, hv.y * f);
    atomicAdd(ap + 2, hv.z * f);
    atomicAdd(ap + 3, hv.w * f);
}

// ---------------------------------------------------------------------------
// interaction: h += BN( softplus(agg@iW1 + ib1) @ iW2 + ib2 )
// 256 threads = 8 waves; each wave owns a 16-row tile (128 rows / block).
// fp32 WMMA 16x16x4, M=16 N=64 K=64 per GEMM, fully unrolled:
// 2 GEMMs * 16 kk * 4 nt = 128 v_wmma per wave.
// Weights staged in LDS pre-swizzled to fragment-major order so each lane's
// B fragment is one aligned ds_load_b64 (no VGPR repacking movs, 0 bank
// conflicts: lane i hits banks 2i,2i+1).
// ---------------------------------------------------------------------------
__global__ __launch_bounds__(256) void interact_kernel(
    const float* __restrict__ agg, float* __restrict__ h,
    const float* __restrict__ iW1, const float* __restrict__ ib1,
    const float* __restrict__ iW2, const float* __restrict__ ib2,
    const float* __restrict__ bng, const float* __restrict__ bnb,
    int l, int n)
{
    // 16 kk * 4 nt * 32 lanes * 2 floats = 4096 floats (16 KB), reused W1->W2
    __shared__ __align__(16) float sWs[16 * 4 * 32 * 2];
    __shared__ __align__(16) float sTile[8 * 16 * TSTR];  // per-wave 16x64 tiles
    __shared__ float sB1[HDIM], sB2[HDIM], sG[HDIM], sBt[HDIM];

    const int tid  = threadIdx.x;
    const int wave = tid >> 5;
    const int lane = tid & 31;
    const int half = lane >> 4;
    const int l16  = lane & 15;
    const int rowBase = blockIdx.x * 128 + wave * 16;

    const float* W1 = iW1 + (size_t)l * HDIM * HDIM;
    const float* W2 = iW2 + (size_t)l * HDIM * HDIM;

    if (tid < HDIM) {
        sB1[tid] = ib1[l * HDIM + tid];
        sB2[tid] = ib2[l * HDIM + tid];
        sG[tid]  = bng[l * HDIM + tid] * BNINV;
        sBt[tid] = bnb[l * HDIM + tid];
    }
    // swizzled fill of W1: pair i -> {W[k0][col], W[k0+1][col]}
    for (int i = tid; i < 2048; i += 256) {
        int li = i & 31;
        int nt = (i >> 5) & 3;
        int kk = i >> 7;
        int k0  = kk * 4 + 2 * (li >> 4);
        int col = nt * 16 + (li & 15);
        sWs[2 * i]     = W1[k0 * HDIM + col];
        sWs[2 * i + 1] = W1[(k0 + 1) * HDIM + col];
    }
    // each wave fills its own 16x64 tile of agg (zero-padded past n)
    for (int i = lane; i < 16 * 16; i += 32) {
        int r = i >> 4, c4 = i & 15;
        int grow = rowBase + r;
        float4 v = make_float4(0.f, 0.f, 0.f, 0.f);
        if (grow < n) v = ((const float4*)(agg + (size_t)grow * HDIM))[c4];
        *(float4*)&sTile[(wave * 16 + r) * TSTR + c4 * 4] = v;
    }
    __syncthreads();

    const float* myA = &sTile[(wave * 16 + l16) * TSTR];
    const v2f*   bw  = (const v2f*)sWs;

    // ---- GEMM1: T1 = aggTile @ W1 + b1 ----
    v8f acc[4];
#pragma unroll
    for (int nt = 0; nt < 4; ++nt) {
        float b = sB1[nt * 16 + l16];
#pragma unroll
        for (int r = 0; r < 8; ++r) acc[nt][r] = b;
    }
#pragma unroll
    for (int kk = 0; kk < 16; ++kk) {
        int k0 = kk * 4 + 2 * half;
        v2f a = *(const v2f*)&myA[k0];
#pragma unroll
        for (int nt = 0; nt < 4; ++nt) {
            v2f b = bw[(kk * 4 + nt) * 32 + lane];
            acc[nt] = __builtin_amdgcn_wmma_f32_16x16x4_f32(
                false, a, false, b, (short)0, acc[nt], false, false);
        }
    }
    __syncthreads();

    // softplus(T1) back into sTile; swap weight buffer to swizzled W2
#pragma unroll
    for (int nt = 0; nt < 4; ++nt)
#pragma unroll
        for (int r = 0; r < 8; ++r) {
            int m = r + 8 * half;
            sTile[(wave * 16 + m) * TSTR + nt * 16 + l16] = softplusf(acc[nt][r]);
        }
    for (int i = tid; i < 2048; i += 256) {
        int li = i & 31;
        int nt = (i >> 5) & 3;
        int kk = i >> 7;
        int k0  = kk * 4 + 2 * (li >> 4);
        int col = nt * 16 + (li & 15);
        sWs[2 * i]     = W2[k0 * HDIM + col];
        sWs[2 * i + 1] = W2[(k0 + 1) * HDIM + col];
    }
    __syncthreads();

    // ---- GEMM2: out = T1 @ W2 + b2 ----
#pragma unroll
    for (int nt = 0; nt < 4; ++nt) {
        float b = sB2[nt * 16 + l16];
#pragma unroll
        for (int r = 0; r < 8; ++r) acc[nt][r] = b;
    }
#pragma unroll
    for (int kk = 0; kk < 16; ++kk) {
        int k0 = kk * 4 + 2 * half;
        v2f a = *(const v2f*)&myA[k0];
#pragma unroll
        for (int nt = 0; nt < 4; ++nt) {
            v2f b = bw[(kk * 4 + nt) * 32 + lane];
            acc[nt] = __builtin_amdgcn_wmma_f32_16x16x4_f32(
                false, a, false, b, (short)0, acc[nt], false, false);
        }
    }

    // epilogue: h += out*inv*gamma + beta
#pragma unroll
    for (int nt = 0; nt < 4; ++nt) {
        int col = nt * 16 + l16;
        float g = sG[col], bt = sBt[col];
#pragma unroll
        for (int r = 0; r < 8; ++r) {
            int grow = rowBase + r + 8 * half;
            if (grow < n) {
                float* p = h + (size_t)grow * HDIM + col;
                *p = *p + acc[nt][r] * g + bt;
            }
        }
    }
}

// ---------------------------------------------------------------------------
// reduce: gsum[j] += sum_n h[n][j]
// ---------------------------------------------------------------------------
__global__ __launch_bounds__(256) void reduce_kernel(
    const float* __restrict__ h, float* __restrict__ gsum, int n)
{
    __shared__ float s[256];
    int tid = threadIdx.x;
    int col = tid & 63;
    int rg  = tid >> 6;
    float acc = 0.0f;
    for (int node = blockIdx.x * 4 + rg; node < n; node += gridDim.x * 4)
        acc += h[(size_t)node * HDIM + col];
    s[tid] = acc;
    __syncthreads();
    if (tid < 64)
        atomicAdd(&gsum[tid], s[tid] + s[tid + 64] + s[tid + 128] + s[tid + 192]);
}

// ---------------------------------------------------------------------------
// final readout MLP -> out[3]
// ---------------------------------------------------------------------------
__global__ __launch_bounds__(64) void final_kernel(
    const float* __restrict__ gsum,
    const float* __restrict__ oW1, const float* __restrict__ ob1,
    const float* __restrict__ og1, const float* __restrict__ obt1,
    const float* __restrict__ oW2, const float* __restrict__ ob2,
    const float* __restrict__ og2, const float* __restrict__ obt2,
    const float* __restrict__ finW, const float* __restrict__ finb,
    float* __restrict__ out, float invN)
{
    __shared__ float sg[64], s1[32], s2[32];
    int t = threadIdx.x;
    sg[t] = gsum[t] * invN;
    __syncthreads();
    if (t < 32) {
        float a = ob1[t];
        for (int k = 0; k < 64; ++k) a = fmaf(sg[k], oW1[k * 32 + t], a);
        s1[t] = softplusf(a) * BNINV * og1[t] + obt1[t];
    }
    __syncthreads();
    if (t < 32) {
        float a = ob2[t];
        for (int k = 0; k < 32; ++k) a = fmaf(s1[k], oW2[k * 32 + t], a);
        s2[t] = softplusf(a) * BNINV * og2[t] + obt2[t];
    }
    __syncthreads();
    if (t < 3) {
        float a = finb[t];
        for (int k = 0; k < 32; ++k) a = fmaf(s2[k], finW[k * 3 + t], a);
        out[t] = a;
    }
}

// ---------------------------------------------------------------------------
extern "C" void kernel_launch(void* const* d_in, const int* in_sizes, int n_in,
                              void* d_out, int out_size, void* d_ws, size_t ws_size,
                              hipStream_t stream) {
    const float* x     = (const float*)d_in[0];
    const int*   ei    = (const int*)  d_in[1];
    const float* dist  = (const float*)d_in[2];
    /* d_in[3] = edge_attr : unused by reference */
    const float* emb_W = (const float*)d_in[4];
    const float* emb_b = (const float*)d_in[5];
    const float* fW1   = (const float*)d_in[6];
    const float* fb1   = (const float*)d_in[7];
    const float* fW2   = (const float*)d_in[8];
    const float* fb2   = (const float*)d_in[9];
    const float* iW1   = (const float*)d_in[10];
    const float* ib1   = (const float*)d_in[11];
    const float* iW2   = (const float*)d_in[12];
    const float* ib2   = (const float*)d_in[13];
    const float* bng   = (const float*)d_in[14];
    const float* bnb   = (const float*)d_in[15];
    const float* oW1   = (const float*)d_in[16];
    const float* ob1   = (const float*)d_in[17];
    const float* og1   = (const float*)d_in[18];
    const float* obt1  = (const float*)d_in[19];
    const float* oW2   = (const float*)d_in[20];
    const float* ob2   = (const float*)d_in[21];
    const float* og2   = (const float*)d_in[22];
    const float* obt2  = (const float*)d_in[23];
    const float* finW  = (const float*)d_in[24];
    const float* finb  = (const float*)d_in[25];
    float* out = (float*)d_out;

    const int N = in_sizes[0] / 16;   // 100000
    const int E = in_sizes[2];        // 1600000

    float* ws     = (float*)d_ws;
    float* h      = ws;
    float* agg    = h   + (size_t)N * HDIM;
    float* fsb    = agg + (size_t)N * HDIM;
    float* w2sum  = fsb + (size_t)LNUM * E;
    float* fb2sum = w2sum + LNUM * HDIM;
    float* gsum   = fb2sum + 4;

    prep_kernel<<<1, 256, 0, stream>>>(fW2, fb2, w2sum, fb2sum, gsum);
    emb_kernel<<<(N + 255) / 256, 256, 0, stream>>>(x, emb_W, emb_b, h, N);
    fs_kernel<<<(E + 255) / 256, 256, 0, stream>>>(dist, fW1, fb1, w2sum, fb2sum, fsb, E);

    for (int l = 0; l < LNUM; ++l) {
        long long aggN = (long long)N * HDIM;
        zero_kernel<<<(int)((aggN + 255) / 256), 256, 0, stream>>>(agg, aggN);
        long long st = (long long)E * 16;
        scatter_kernel<<<(int)((st + 255) / 256), 256, 0, stream>>>(
            ei, h, fsb + (size_t)l * E, agg, E);
        interact_kernel<<<(N + 127) / 128, 256, 0, stream>>>(
            agg, h, iW1, ib1, iW2, ib2, bng, bnb, l, N);
    }

    reduce_kernel<<<512, 256, 0, stream>>>(h, gsum, N);
    final_kernel<<<1, 64, 0, stream>>>(gsum, oW1, ob1, og1, obt1,
                                       oW2, ob2, og2, obt2, finW, finb,
                                       out, 1.0f / (float)N);
}